// selected_meanPool_37838661878408
// MI455X (gfx1250) — compile-verified
//
#include <hip/hip_runtime.h>
#include <hip/hip_bf16.h>

// Masked dilated mean-pool, shapes fixed by the reference:
//   x:    (B=8, C=64, H=256, W=256) fp32
//   mask: (B=8, 1,    H=256, W=256) fp32
//   K=5, DILATE=2, PAD=4, reflect padding.
#define B_   8
#define C_   64
#define H_   256
#define W_   256
#define K_   5
#define DIL_ 2
#define PAD_ 4
#define NC_  2            // channels per pipeline iteration
#define NIT_ (C_ / NC_)   // 32 iterations
#define NLD_ (NC_ * K_)   // 10 async b32 loads per thread per iteration

#if __has_builtin(__builtin_amdgcn_global_load_async_to_lds_b32)
#define HAVE_ASYNC_LDS 1
#endif

typedef __attribute__((address_space(1))) int* gint_ptr;  // global (device) AS
typedef __attribute__((address_space(3))) int* lint_ptr;  // LDS (shared) AS

__device__ __forceinline__ void async_copy_b32(const float* g, float* l) {
#ifdef HAVE_ASYNC_LDS
  // gfx1250: GLOBAL_LOAD_ASYNC_TO_LDS_B32 (tracked by ASYNCcnt)
  __builtin_amdgcn_global_load_async_to_lds_b32(
      (gint_ptr)(g), (lint_ptr)(l), /*imm offset=*/0, /*cpol=*/0);
#else
  *l = *g;  // synchronous fallback (waits below become no-ops)
#endif
}

__device__ __forceinline__ void async_wait_all() {
#if __has_builtin(__builtin_amdgcn_s_wait_asynccnt)
  __builtin_amdgcn_s_wait_asynccnt(0);
#else
  asm volatile("s_wait_asynccnt 0" ::: "memory");
#endif
}

__device__ __forceinline__ void async_wait_batch() {
  // <= NLD_ outstanding: the batch just issued remains, previous batch retired
  // (async loads complete in order per the CDNA5 ISA).
#if __has_builtin(__builtin_amdgcn_s_wait_asynccnt)
  __builtin_amdgcn_s_wait_asynccnt(NLD_);
#else
  asm volatile("s_wait_asynccnt 10" ::: "memory");
#endif
}

// jnp.pad mode="reflect" index mapping for a 256-long axis.
__device__ __forceinline__ int reflect256(int v) {
  v = (v < 0) ? -v : v;
  v = (v > 255) ? (510 - v) : v;
  return v;
}

__global__ __launch_bounds__(W_) void selected_meanpool_kernel(
    const float* __restrict__ x, const float* __restrict__ mask,
    float* __restrict__ out) {
  __shared__ float smMask[K_][W_];         //  5 KB
  __shared__ float smX[3][NLD_][W_];       // 30 KB, triple-buffered

  const int t  = threadIdx.x;  // column 0..255
  const int y  = blockIdx.x;   // output row
  const int bb = blockIdx.y;   // batch

  // Reflected source rows for the 5 vertical taps (dy = -4,-2,0,2,4).
  int rrow[K_];
#pragma unroll
  for (int i = 0; i < K_; ++i) rrow[i] = reflect256(y + i * DIL_ - PAD_);

  const float* mbase = mask + (size_t)bb * (H_ * W_);
  const float* xbase = x + (size_t)bb * C_ * H_ * W_;

  // Issue the NC_ channels of iteration `it` into LDS buffer `buf`.
  auto issue_iter = [&](int it, int buf) {
#pragma unroll
    for (int cc = 0; cc < NC_; ++cc) {
      const float* xc = xbase + (size_t)(it * NC_ + cc) * (H_ * W_);
#pragma unroll
      for (int i = 0; i < K_; ++i)
        async_copy_b32(xc + (size_t)rrow[i] * W_ + t, &smX[buf][cc * K_ + i][t]);
    }
  };

  // Prologue: stage iteration 0 (buffer 0) + the 5 mask rows.
  issue_iter(0, 0);
#pragma unroll
  for (int i = 0; i < K_; ++i)
    async_copy_b32(mbase + (size_t)rrow[i] * W_ + t, &smMask[i][t]);
  async_wait_all();
  __syncthreads();

  // Per-pixel match pattern (channel independent): 25 bits + 1/count.
  int cols[K_];
#pragma unroll
  for (int j = 0; j < K_; ++j) cols[j] = reflect256(t + j * DIL_ - PAD_);

  const float m0 = smMask[K_ / 2][t];  // dy == 0 row, own column
  unsigned bits = 0u;
#pragma unroll
  for (int i = 0; i < K_; ++i) {
#pragma unroll
    for (int j = 0; j < K_; ++j) {
      bits |= (smMask[i][cols[j]] == m0 ? 1u : 0u) << (i * K_ + j);
    }
  }
  const float invCnt = 1.0f / (float)__popc(bits);  // center always matches

  // Pipelined channel loop: triple buffer, ONE barrier per iteration.
  // Buffer written by issue(it+1) was last read at compute(it-2), which
  // precedes barrier(it-1) in program order for every wave -> safe.
  for (int it = 0; it < NIT_; ++it) {
    if (it + 1 < NIT_) {
      issue_iter(it + 1, (it + 1) % 3);
      async_wait_batch();  // retire this iteration's loads
    } else {
      async_wait_all();
    }
    __syncthreads();  // all waves' loads for buf[it%3] have landed

    const float(*buf)[W_] = smX[it % 3];
#pragma unroll
    for (int cc = 0; cc < NC_; ++cc) {
      float acc = 0.0f;
      unsigned m = bits;
#pragma unroll
      for (int i = 0; i < K_; ++i) {
#pragma unroll
        for (int j = 0; j < K_; ++j) {
          const float v = buf[cc * K_ + i][cols[j]];
          acc += (m & 1u) ? v : 0.0f;  // v_cndmask + v_add
          m >>= 1;
        }
      }
      const int c = it * NC_ + cc;
      float* op = out + (((size_t)bb * C_ + c) * H_ + y) * W_ + t;
      __builtin_nontemporal_store(acc * invCnt, op);  // stream out, keep x in L2
    }
  }
}

extern "C" void kernel_launch(void* const* d_in, const int* in_sizes, int n_in,
                              void* d_out, int out_size, void* d_ws,
                              size_t ws_size, hipStream_t stream) {
  (void)in_sizes; (void)n_in; (void)out_size; (void)d_ws; (void)ws_size;
  const float* x    = (const float*)d_in[0];
  const float* mask = (const float*)d_in[1];
  float* out        = (float*)d_out;

  dim3 grid(H_, B_);  // one block per (row, batch)
  dim3 block(W_);     // 256 threads = 8 wave32s
  selected_meanpool_kernel<<<grid, block, 0, stream>>>(x, mask, out);
}